// BktModel_34050500722875
// MI455X (gfx1250) — compile-verified
//
#include <hip/hip_runtime.h>
#include <stdint.h>

#define BKT_T   500
#define BKT_KCS 50

// Numerically stable 2-way logsumexp.
__device__ __forceinline__ float lse2f(float a, float b) {
    float m = fmaxf(a, b);
    return m + __logf(__expf(a - m) + __expf(b - m));
}

__global__ __launch_bounds__(64, 1) void bkt_forward_kernel(
    const int*   __restrict__ corr,               // [B,T]
    const int*   __restrict__ kc,                 // [B,T]
    const int*   __restrict__ problem,            // [B,T]
    const float* __restrict__ trans_logits,       // [C,2,2]
    const float* __restrict__ obs_logits_problem, // [P,2,2]
    const float* __restrict__ obs_logits_kc,      // [C,2,2]
    const float* __restrict__ init_logits,        // [C,2]
    float*       __restrict__ out,                // [B,T,2]
    int T)
{
    __shared__ int s_corr[BKT_T];
    __shared__ int s_kc[BKT_T];
    __shared__ int s_prob[BKT_T];

    const int b   = blockIdx.x;
    const int tid = threadIdx.x;
    const long long base = (long long)b * T;

#if defined(__gfx1250__)
    // CDNA5 async global->LDS staging (ASYNCcnt-tracked DMA-style copies).
    for (int t = tid; t < T; t += 64) {
        const int* g0 = corr    + base + t;
        const int* g1 = kc      + base + t;
        const int* g2 = problem + base + t;
        uint32_t l0 = (uint32_t)(uintptr_t)(&s_corr[t]); // generic shared addr: low 32 bits = LDS offset
        uint32_t l1 = (uint32_t)(uintptr_t)(&s_kc[t]);
        uint32_t l2 = (uint32_t)(uintptr_t)(&s_prob[t]);
        asm volatile("global_load_async_to_lds_b32 %0, %1, off" :: "v"(l0), "v"(g0) : "memory");
        asm volatile("global_load_async_to_lds_b32 %0, %1, off" :: "v"(l1), "v"(g1) : "memory");
        asm volatile("global_load_async_to_lds_b32 %0, %1, off" :: "v"(l2), "v"(g2) : "memory");
    }
    asm volatile("s_wait_asynccnt 0" ::: "memory");
#else
    for (int t = tid; t < T; t += 64) {
        s_corr[t] = corr[base + t];
        s_kc[t]   = kc[base + t];
        s_prob[t] = problem[base + t];
    }
#endif
    __syncthreads();

    // Warm WGP$/L2 for the data-dependent obs_logits_problem gather
    // (emits global_prefetch_b8 on gfx1250).
    for (int t = tid; t < T; t += 64) {
        __builtin_prefetch(&obs_logits_problem[(size_t)s_prob[t] * 4], 0, 3);
    }

    const int c = tid;                 // one 2-state HMM chain per thread
    if (c >= BKT_KCS) return;

    // log_softmax(trans_logits[c], axis=0): normalize over DEST state (rows i, fixed col j)
    const float4 tv  = *reinterpret_cast<const float4*>(&trans_logits[c * 4]); // t[i][j]
    const float nj0  = lse2f(tv.x, tv.z);   // column j=0
    const float nj1  = lse2f(tv.y, tv.w);   // column j=1
    const float lt00 = tv.x - nj0, lt01 = tv.y - nj1;
    const float lt10 = tv.z - nj0, lt11 = tv.w - nj1;

    const float4 okv = *reinterpret_cast<const float4*>(&obs_logits_kc[c * 4]); // ok[s][o]
    const float ok00 = okv.x, ok01 = okv.y, ok10 = okv.z, ok11 = okv.w;

    const float i0 = init_logits[c * 2 + 0];
    const float i1 = init_logits[c * 2 + 1];
    const float ni = lse2f(i0, i1);
    float a0 = i0 - ni, a1 = i1 - ni;   // log_alpha for this (b, c) chain

    for (int t = 0; t < T; ++t) {
        if (s_kc[t] != c) continue;     // exactly one chain per (b,t) is active
        const int y = s_corr[t];
        const int p = s_prob[t];

        const float4 op = *reinterpret_cast<const float4*>(
            &obs_logits_problem[(size_t)p * 4]);        // op[s][o]

        // logit = obs_problem[p] + obs_kc[c]; log_softmax over outputs o
        const float l00 = op.x + ok00, l01 = op.y + ok01;
        const float l10 = op.z + ok10, l11 = op.w + ok11;
        const float z0  = lse2f(l00, l01);
        const float z1  = lse2f(l10, l11);
        const float lo00 = l00 - z0, lo01 = l01 - z0;   // log_obs[s][o]
        const float lo10 = l10 - z1, lo11 = l11 - z1;

        // predictive distribution over outputs, normalized
        const float py0 = lse2f(lo00 + a0, lo10 + a1);
        const float py1 = lse2f(lo01 + a0, lo11 + a1);
        const float zn  = lse2f(py0, py1);
        *reinterpret_cast<float2*>(&out[(base + t) * 2]) = make_float2(py0 - zn, py1 - zn);

        // alpha update: new_a[i] = lse_j( log_obs[j][y] + a[j] + log_t[i][j] )
        const float ly0 = (y == 0) ? lo00 : lo01;       // log_obs[state0][y]
        const float ly1 = (y == 0) ? lo10 : lo11;       // log_obs[state1][y]
        const float e0  = ly0 + a0;
        const float e1  = ly1 + a1;
        const float na0 = lse2f(e0 + lt00, e1 + lt01);
        const float na1 = lse2f(e0 + lt10, e1 + lt11);
        a0 = na0; a1 = na1;
    }
}

extern "C" void kernel_launch(void* const* d_in, const int* in_sizes, int n_in,
                              void* d_out, int out_size, void* d_ws, size_t ws_size,
                              hipStream_t stream) {
    (void)n_in; (void)d_ws; (void)ws_size; (void)out_size;
    const int*   corr  = (const int*)d_in[0];
    const int*   kcp   = (const int*)d_in[1];
    const int*   prob  = (const int*)d_in[2];
    const float* trans = (const float*)d_in[3];
    const float* obs_p = (const float*)d_in[4];
    const float* obs_k = (const float*)d_in[5];
    const float* initl = (const float*)d_in[6];
    float* out = (float*)d_out;

    const int T = BKT_T;
    const int B = in_sizes[0] / T;   // 1024

    bkt_forward_kernel<<<dim3(B), dim3(64), 0, stream>>>(
        corr, kcp, prob, trans, obs_p, obs_k, initl, out, T);
}